// InvariantPointAttention_40896678592840
// MI455X (gfx1250) — compile-verified
//
#include <hip/hip_runtime.h>
#include <hip/hip_bf16.h>

#define NRES 768
#define SD   384
#define PDIM 128
#define DH   16
#define NH   12
#define NP   4
#define NPV  8
#define FEAT 2112            // NH*(PDIM+DH+NPV*3+NPV)
#define SQRT_INV_D 0.25f
#define W_C 0.23570226039551584f   // sqrt(2/(9*4))
#define W_L 0.5773502691896258f    // sqrt(1/3)

typedef __attribute__((ext_vector_type(2))) float v2f;
typedef __attribute__((ext_vector_type(8))) float v8f;

__device__ __forceinline__ v8f wmma4(v2f a, v2f b, v8f c) {
  // D = A(16x4) * B(4x16) + C, fp32, wave32
  return __builtin_amdgcn_wmma_f32_16x16x4_f32(false, a, false, b, (short)0, c, false, false);
}
__device__ __forceinline__ float wsum(float x) {
#pragma unroll
  for (int o = 16; o > 0; o >>= 1) x += __shfl_xor(x, o, 32);
  return x;
}
__device__ __forceinline__ float wmax(float x) {
#pragma unroll
  for (int o = 16; o > 0; o >>= 1) x = fmaxf(x, __shfl_xor(x, o, 32));
  return x;
}

// ---------------------------------------------------------------------------
// K0: projections + frame application. One block per residue n.
// Layouts: qh/kh/vh [H][N][16], qg/kg [H][N][16] (12 used, pad 0),
//          vg [H][N][32] (24 used, pad 0), qsq/ksq [H][N]
// ---------------------------------------------------------------------------
__global__ void __launch_bounds__(256)
proj_kernel(const float* __restrict__ single, const float* __restrict__ rotations,
            const float* __restrict__ translations,
            const float* __restrict__ Wqkv, const float* __restrict__ Wqk_pts,
            const float* __restrict__ Wv_pts,
            float* __restrict__ qh, float* __restrict__ kh, float* __restrict__ vh,
            float* __restrict__ qg, float* __restrict__ kg, float* __restrict__ vg,
            float* __restrict__ qsq, float* __restrict__ ksq)
{
  const int n = blockIdx.x;
  const int tid = threadIdx.x;
  __shared__ float srow[SD];
  __shared__ float qkv_s[3 * NH * DH];   // 576
  __shared__ float qkp_s[2 * NH * NP * 3]; // 288
  __shared__ float vp_s[NH * NPV * 3];   // 288
  __shared__ float qgS[NH * NP * 3];     // 144
  __shared__ float kgS[NH * NP * 3];
  __shared__ float vgS[NH * NPV * 3];    // 288

  for (int c = tid; c < SD; c += 256) srow[c] = single[n * SD + c];
  __syncthreads();

  for (int o = tid; o < 576; o += 256) {
    float s = 0.f;
    for (int k2 = 0; k2 < SD; ++k2) s += srow[k2] * Wqkv[k2 * 576 + o];
    qkv_s[o] = s;
  }
  for (int o = tid; o < 288; o += 256) {
    float s = 0.f, s2 = 0.f;
    for (int k2 = 0; k2 < SD; ++k2) {
      float x = srow[k2];
      s  += x * Wqk_pts[k2 * 288 + o];
      s2 += x * Wv_pts[k2 * 288 + o];
    }
    qkp_s[o] = s;
    vp_s[o]  = s2;
  }
  __syncthreads();

  const float* R  = rotations + n * 9;
  const float* tr = translations + n * 3;

  // global-frame q/k points: qg_b = R[b,a]*pt_a + t_b
  for (int idx = tid; idx < NH * NP; idx += 256) {
    int h = idx / NP, p = idx % NP;
    float lq[3], lk[3];
#pragma unroll
    for (int t = 0; t < 3; ++t) {
      lq[t] = qkp_s[p * 36 + t * 12 + h];
      lk[t] = qkp_s[144 + p * 36 + t * 12 + h];
    }
#pragma unroll
    for (int b = 0; b < 3; ++b) {
      qgS[h * 12 + p * 3 + b] = R[b*3+0]*lq[0] + R[b*3+1]*lq[1] + R[b*3+2]*lq[2] + tr[b];
      kgS[h * 12 + p * 3 + b] = R[b*3+0]*lk[0] + R[b*3+1]*lk[1] + R[b*3+2]*lk[2] + tr[b];
    }
  }
  for (int idx = tid; idx < NH * NPV; idx += 256) {
    int h = idx / NPV, p = idx % NPV;
    float lv[3];
#pragma unroll
    for (int t = 0; t < 3; ++t) lv[t] = vp_s[p * 36 + t * 12 + h];
#pragma unroll
    for (int b = 0; b < 3; ++b)
      vgS[h * 24 + p * 3 + b] = R[b*3+0]*lv[0] + R[b*3+1]*lv[1] + R[b*3+2]*lv[2] + tr[b];
  }
  __syncthreads();

  // scatter q/k/v:  q[n,d,h] = qkv[n, d*H + h]
  for (int idx = tid; idx < NH * DH; idx += 256) {
    int d = idx / NH, h = idx % NH;
    qh[(h * NRES + n) * DH + d] = qkv_s[idx];
    kh[(h * NRES + n) * DH + d] = qkv_s[192 + idx];
    vh[(h * NRES + n) * DH + d] = qkv_s[384 + idx];
  }
  for (int idx = tid; idx < NH * 16; idx += 256) {
    int h = idx / 16, c = idx % 16;
    qg[(h * NRES + n) * 16 + c] = (c < 12) ? qgS[h * 12 + c] : 0.f;
    kg[(h * NRES + n) * 16 + c] = (c < 12) ? kgS[h * 12 + c] : 0.f;
  }
  for (int idx = tid; idx < NH * 32; idx += 256) {
    int h = idx / 32, c = idx % 32;
    vg[(h * NRES + n) * 32 + c] = (c < 24) ? vgS[h * 24 + c] : 0.f;
  }
  if (tid < NH) {
    float sq = 0.f, sk = 0.f;
    for (int c = 0; c < 12; ++c) {
      float a = qgS[tid * 12 + c]; sq += a * a;
      float b = kgS[tid * 12 + c]; sk += b * b;
    }
    qsq[tid * NRES + n] = sq;
    ksq[tid * NRES + n] = sk;
  }
}

// ---------------------------------------------------------------------------
// K1: b = pair_rep @ Wb -> logits[h][i][j]. One wave per (i,j); first pass
// over pair_rep (302 MB), each row read once, all 12 heads computed from it.
// ---------------------------------------------------------------------------
__global__ void __launch_bounds__(256)
bias_kernel(const float* __restrict__ pair, const float* __restrict__ Wb,
            float* __restrict__ logits)
{
  __shared__ float WbS[PDIM * NH];
  const int tid = threadIdx.x;
  for (int c = tid; c < PDIM * NH; c += 256) WbS[c] = Wb[c];
  __syncthreads();

  const int wave = tid >> 5, lane = tid & 31;
  const size_t pi = (size_t)blockIdx.x * 8 + wave;
  const int i = (int)(pi / NRES), j = (int)(pi % NRES);
  const float* pr = pair + pi * PDIM;

  float part[NH];
#pragma unroll
  for (int h = 0; h < NH; ++h) part[h] = 0.f;
#pragma unroll
  for (int cc = 0; cc < 4; ++cc) {
    int c = lane + 32 * cc;
    float x = pr[c];
    const float* wb = &WbS[c * NH];
#pragma unroll
    for (int h = 0; h < NH; ++h) part[h] += x * wb[h];
  }
  float keep = 0.f;
#pragma unroll
  for (int h = 0; h < NH; ++h) {
    float s = wsum(part[h]);
    if (lane == h) keep = s;
  }
  if (lane < NH) logits[((size_t)lane * NRES + i) * NRES + j] = keep;
}

// ---------------------------------------------------------------------------
// K2: per (head, 16-row i tile): WMMA QK^T + point-cross, combine with bias &
// distance terms, softmax over j in LDS, overwrite logits with a.
// ---------------------------------------------------------------------------
__global__ void __launch_bounds__(256)
logits_softmax_kernel(const float* __restrict__ qh, const float* __restrict__ kh,
                      const float* __restrict__ qg, const float* __restrict__ kg,
                      const float* __restrict__ qsq, const float* __restrict__ ksq,
                      const float* __restrict__ gamma, float* __restrict__ logits)
{
  __shared__ float lrow[16][NRES];  // 48 KB of the 320 KB/WGP LDS
  const int h  = blockIdx.x / 48;
  const int i0 = (blockIdx.x % 48) * 16;
  const int tid = threadIdx.x, wave = tid >> 5, lane = tid & 31;
  const int g = lane >> 4, ml = lane & 15;
  const float scale = gamma[h] * (W_C * 0.5f);

  v2f aq[4], ag[4];
#pragma unroll
  for (int kk = 0; kk < 4; ++kk) {
    aq[kk] = *(const v2f*)(qh + ((size_t)(h * NRES + i0 + ml)) * DH + kk * 4 + 2 * g);
    ag[kk] = *(const v2f*)(qg + ((size_t)(h * NRES + i0 + ml)) * 16 + kk * 4 + 2 * g);
  }

  for (int jt = wave; jt < 48; jt += 8) {
    const int j0 = jt * 16;
    v8f cdp = {}; v8f cx = {};
#pragma unroll
    for (int kk = 0; kk < 4; ++kk) {
      v2f bq = *(const v2f*)(kh + ((size_t)(h * NRES + j0 + ml)) * DH + kk * 4 + 2 * g);
      cdp = wmma4(aq[kk], bq, cdp);
      v2f bg = *(const v2f*)(kg + ((size_t)(h * NRES + j0 + ml)) * 16 + kk * 4 + 2 * g);
      cx = wmma4(ag[kk], bg, cx);
    }
#pragma unroll
    for (int r = 0; r < 8; ++r) {
      const int row = r + 8 * g;
      const int i = i0 + row, j = j0 + ml;
      float bij  = logits[((size_t)h * NRES + i) * NRES + j];
      float dist = qsq[h * NRES + i] + ksq[h * NRES + j] - 2.f * cx[r];
      lrow[row][j] = W_L * (SQRT_INV_D * cdp[r] + bij - scale * dist);
    }
  }
  __syncthreads();

  for (int row = wave; row < 16; row += 8) {
    float m = -3.402823e38f;
    for (int j = lane; j < NRES; j += 32) m = fmaxf(m, lrow[row][j]);
    m = wmax(m);
    float s = 0.f;
    for (int j = lane; j < NRES; j += 32) {
      float e = __expf(lrow[row][j] - m);
      lrow[row][j] = e;
      s += e;
    }
    s = wsum(s);
    const float inv = 1.f / s;
    float* dst = logits + ((size_t)h * NRES + i0 + row) * NRES;
    for (int j = lane; j < NRES; j += 32) dst[j] = lrow[row][j] * inv;
  }
}

// ---------------------------------------------------------------------------
// K3: out_value = a@V, op = a@Vg (WMMA, 8-wave K split + LDS reduce), then
// inverse frames + norms into feat.
// ---------------------------------------------------------------------------
__global__ void __launch_bounds__(256)
value_points_kernel(const float* __restrict__ a, const float* __restrict__ vh,
                    const float* __restrict__ vg, const float* __restrict__ rotations,
                    const float* __restrict__ translations, float* __restrict__ feat)
{
  __shared__ float red[8][16][16];
  __shared__ float outv[16][16];
  __shared__ float outp[16][32];
  const int h  = blockIdx.x / 48;
  const int i0 = (blockIdx.x % 48) * 16;
  const int tid = threadIdx.x, wave = tid >> 5, lane = tid & 31;
  const int g = lane >> 4, ml = lane & 15;

  v8f c0 = {}, c1 = {}, c2 = {};
  for (int jt = wave; jt < 48; jt += 8) {
    const int j0 = jt * 16;
#pragma unroll
    for (int kk = 0; kk < 4; ++kk) {
      v2f af = *(const v2f*)(a + ((size_t)h * NRES + i0 + ml) * NRES + j0 + kk * 4 + 2 * g);
      const int krow = j0 + kk * 4 + 2 * g;
      const float* v0 = vh + ((size_t)(h * NRES + krow)) * DH + ml;
      v2f b0 = { v0[0], v0[DH] };
      c0 = wmma4(af, b0, c0);
      const float* v1 = vg + ((size_t)(h * NRES + krow)) * 32 + ml;
      v2f b1 = { v1[0], v1[32] };
      c1 = wmma4(af, b1, c1);
      v2f b2 = { v1[16], v1[48] };
      c2 = wmma4(af, b2, c2);
    }
  }

  const int row = tid >> 4, col = tid & 15;
#pragma unroll
  for (int r = 0; r < 8; ++r) red[wave][r + 8 * g][ml] = c0[r];
  __syncthreads();
  { float s = 0.f;
#pragma unroll
    for (int w2 = 0; w2 < 8; ++w2) s += red[w2][row][col];
    outv[row][col] = s; }
  __syncthreads();
#pragma unroll
  for (int r = 0; r < 8; ++r) red[wave][r + 8 * g][ml] = c1[r];
  __syncthreads();
  { float s = 0.f;
#pragma unroll
    for (int w2 = 0; w2 < 8; ++w2) s += red[w2][row][col];
    outp[row][col] = s; }
  __syncthreads();
#pragma unroll
  for (int r = 0; r < 8; ++r) red[wave][r + 8 * g][ml] = c2[r];
  __syncthreads();
  { float s = 0.f;
#pragma unroll
    for (int w2 = 0; w2 < 8; ++w2) s += red[w2][row][col];
    outp[row][16 + col] = s; }
  __syncthreads();

  {
    const int i = i0 + row;
    feat[(size_t)i * FEAT + 1536 + col * NH + h] = outv[row][col];
  }
  if (tid < 128) {
    const int prow = tid >> 3, p = tid & 7;
    const int i = i0 + prow;
    const float* R  = rotations + i * 9;
    const float* tr = translations + i * 3;
    float x0 = outp[prow][p * 3 + 0] - tr[0];
    float x1 = outp[prow][p * 3 + 1] - tr[1];
    float x2 = outp[prow][p * 3 + 2] - tr[2];
    float o[3];
#pragma unroll
    for (int b = 0; b < 3; ++b) o[b] = R[0*3+b]*x0 + R[1*3+b]*x1 + R[2*3+b]*x2;
    float nrm = sqrtf(o[0]*o[0] + o[1]*o[1] + o[2]*o[2]);
    float* fb = feat + (size_t)i * FEAT;
    fb[1728 + (p * NH + h) * 3 + 0] = o[0];
    fb[1728 + (p * NH + h) * 3 + 1] = o[1];
    fb[1728 + (p * NH + h) * 3 + 2] = o[2];
    fb[2016 + p * NH + h] = nrm;
  }
}

// ---------------------------------------------------------------------------
// K4: out_pair[i] = a[heads x 768] @ pair_rep[i] (768x128). One block per i:
// second (final) pass over pair_rep, each element read exactly once.
// ---------------------------------------------------------------------------
__global__ void __launch_bounds__(256)
pair_out_kernel(const float* __restrict__ a, const float* __restrict__ pair,
                float* __restrict__ feat)
{
  __shared__ float red[8][16][16];
  const int i = blockIdx.x;
  const int tid = threadIdx.x, wave = tid >> 5, lane = tid & 31;
  const int g = lane >> 4, ml = lane & 15;
  const v2f z2 = {0.f, 0.f};

  v8f acc[8];
  v8f z8 = {};
#pragma unroll
  for (int zt = 0; zt < 8; ++zt) acc[zt] = z8;

  for (int jt = wave; jt < 48; jt += 8) {
    const int j0 = jt * 16;
    if (jt + 8 < 48)
      __builtin_prefetch(pair + ((size_t)i * NRES + (jt + 8) * 16) * PDIM + lane * 4, 0, 0);
    v2f af[4];
#pragma unroll
    for (int kk = 0; kk < 4; ++kk) {
      af[kk] = (ml < NH)
        ? *(const v2f*)(a + ((size_t)ml * NRES + i) * NRES + j0 + kk * 4 + 2 * g)
        : z2;
    }
    const float* pb = pair + ((size_t)i * NRES + j0) * PDIM;
#pragma unroll
    for (int zt = 0; zt < 8; ++zt) {
#pragma unroll
      for (int kk = 0; kk < 4; ++kk) {
        const float* bp = pb + (size_t)(kk * 4 + 2 * g) * PDIM + zt * 16 + ml;
        v2f bf = { bp[0], bp[PDIM] };
        acc[zt] = wmma4(af[kk], bf, acc[zt]);
      }
    }
  }

  const int row = tid >> 4, col = tid & 15;
  for (int zt = 0; zt < 8; ++zt) {
#pragma unroll
    for (int r = 0; r < 8; ++r) red[wave][r + 8 * g][ml] = acc[zt][r];
    __syncthreads();
    float s = 0.f;
#pragma unroll
    for (int w2 = 0; w2 < 8; ++w2) s += red[w2][row][col];
    if (row < NH) feat[(size_t)i * FEAT + (zt * 16 + col) * NH + row] = s;
    __syncthreads();
  }
}

// ---------------------------------------------------------------------------
// K5: out = feat [768x2112] @ Wout [2112x384] + bout, WMMA with 8-wave K split
// ---------------------------------------------------------------------------
__global__ void __launch_bounds__(256)
final_kernel(const float* __restrict__ feat, const float* __restrict__ Wout,
             const float* __restrict__ bout, float* __restrict__ out)
{
  __shared__ float red[8][16][16];
  const int i0 = (blockIdx.x / 24) * 16;
  const int c0 = (blockIdx.x % 24) * 16;
  const int tid = threadIdx.x, wave = tid >> 5, lane = tid & 31;
  const int g = lane >> 4, ml = lane & 15;

  v8f acc = {};
  for (int kt = wave; kt < 132; kt += 8) {
    const int kb = kt * 16;
#pragma unroll
    for (int kk = 0; kk < 4; ++kk) {
      v2f af = *(const v2f*)(feat + (size_t)(i0 + ml) * FEAT + kb + kk * 4 + 2 * g);
      const float* wp = Wout + (size_t)(kb + kk * 4 + 2 * g) * SD + c0 + ml;
      v2f bf = { wp[0], wp[SD] };
      acc = wmma4(af, bf, acc);
    }
  }
#pragma unroll
  for (int r = 0; r < 8; ++r) red[wave][r + 8 * g][ml] = acc[r];
  __syncthreads();
  const int row = tid >> 4, col = tid & 15;
  float s = bout[c0 + col];
#pragma unroll
  for (int w2 = 0; w2 < 8; ++w2) s += red[w2][row][col];
  out[(size_t)(i0 + row) * SD + c0 + col] = s;
}

// ---------------------------------------------------------------------------
extern "C" void kernel_launch(void* const* d_in, const int* in_sizes, int n_in,
                              void* d_out, int out_size, void* d_ws, size_t ws_size,
                              hipStream_t stream) {
  (void)in_sizes; (void)n_in; (void)out_size; (void)ws_size;
  const float* single = (const float*)d_in[0];
  const float* pair   = (const float*)d_in[1];
  const float* rot    = (const float*)d_in[2];
  const float* trans  = (const float*)d_in[3];
  const float* Wqkv   = (const float*)d_in[4];
  const float* Wb     = (const float*)d_in[5];
  const float* Wqk    = (const float*)d_in[6];
  const float* Wv     = (const float*)d_in[7];
  const float* gamma  = (const float*)d_in[8];
  const float* Wout   = (const float*)d_in[9];
  const float* bout   = (const float*)d_in[10];
  float* out = (float*)d_out;

  float* w = (float*)d_ws;
  float* qh = w;  w += (size_t)NH * NRES * DH;
  float* kh = w;  w += (size_t)NH * NRES * DH;
  float* vh = w;  w += (size_t)NH * NRES * DH;
  float* qg = w;  w += (size_t)NH * NRES * 16;
  float* kg = w;  w += (size_t)NH * NRES * 16;
  float* vg = w;  w += (size_t)NH * NRES * 32;
  float* qsq = w; w += (size_t)NH * NRES;
  float* ksq = w; w += (size_t)NH * NRES;
  float* logits = w; w += (size_t)NH * NRES * NRES;   // b, then a (28.3 MB)
  float* feat = w;   w += (size_t)NRES * FEAT;        // 6.5 MB

  proj_kernel<<<NRES, 256, 0, stream>>>(single, rot, trans, Wqkv, Wqk, Wv,
                                        qh, kh, vh, qg, kg, vg, qsq, ksq);
  bias_kernel<<<(NRES * NRES) / 8, 256, 0, stream>>>(pair, Wb, logits);
  logits_softmax_kernel<<<NH * 48, 256, 0, stream>>>(qh, kh, qg, kg, qsq, ksq,
                                                     gamma, logits);
  value_points_kernel<<<NH * 48, 256, 0, stream>>>(logits, vh, vg, rot, trans, feat);
  pair_out_kernel<<<NRES, 256, 0, stream>>>(logits, pair, feat);
  final_kernel<<<48 * 24, 256, 0, stream>>>(feat, Wout, bout, out);
}